// LSTM_13915694039521
// MI455X (gfx1250) — compile-verified
//
#include <hip/hip_runtime.h>
#include <math.h>

#define INPD   1024
#define HIDD   2048
#define TSTEPS 1024
#define KAUG   3072   /* INP+HID */
#define ROWS4H 8192   /* 4*HID   */

typedef __attribute__((ext_vector_type(2))) float v2f;
typedef __attribute__((ext_vector_type(8))) float v8f;

// -----------------------------------------------------------------------------
// Kernel A: xz[r][t] = sum_k W[r][k] * x_seq[t][k] + b[r]   (r<8192, t<1024)
// fp32 WMMA 16x16x4; one wave computes a 16(M) x 64(N=time) tile, reusing the
// A fragment across 4 WMMAs. Output layout [8192][1024] -> coalesced stores,
// and the per-step kernel later reads 4 scalars per wave.
// -----------------------------------------------------------------------------
__global__ __launch_bounds__(256)
void lstm_xgemm_wmma(const float* __restrict__ x,   // [T, INP]
                     const float* __restrict__ Wi, const float* __restrict__ Wf,
                     const float* __restrict__ Wo, const float* __restrict__ Wg,
                     const float* __restrict__ bi, const float* __restrict__ bf,
                     const float* __restrict__ bo, const float* __restrict__ bg,
                     float* __restrict__ xz)        // [8192, 1024]
{
    const int wid   = (blockIdx.x * blockDim.x + threadIdx.x) >> 5; // 0..8191
    const int lane  = threadIdx.x & 31;
    const int mtile = wid >> 4;          // 0..511  (16 rows each)
    const int n0    = (wid & 15) * 64;   // 4 consecutive N-tiles of 16

    const int m0   = mtile * 16;         // global row in concatenated W
    const int gate = m0 >> 11;           // tile never straddles a gate (16 | 2048)
    const int jrow = m0 & 2047;
    const float* Wmat = (gate == 0) ? Wi : (gate == 1) ? Wf : (gate == 2) ? Wo : Wg;
    const float* bvec = (gate == 0) ? bi : (gate == 1) ? bf : (gate == 2) ? bo : bg;

    const int ml = lane & 15;            // M row (A) / N col (B,D)
    const int kh = (lane >> 4) << 1;     // K sub-offset: 0 or 2

    const float* arow = Wmat + (size_t)(jrow + ml) * KAUG; // x-part = cols [0,1024)

    v8f acc0 = {}, acc1 = {}, acc2 = {}, acc3 = {};
    for (int k = 0; k < INPD; k += 4) {
        v2f a  = *(const v2f*)(arow + k + kh);
        v2f b0 = *(const v2f*)(x + (size_t)(n0 +  0 + ml) * INPD + k + kh);
        v2f b1 = *(const v2f*)(x + (size_t)(n0 + 16 + ml) * INPD + k + kh);
        v2f b2 = *(const v2f*)(x + (size_t)(n0 + 32 + ml) * INPD + k + kh);
        v2f b3 = *(const v2f*)(x + (size_t)(n0 + 48 + ml) * INPD + k + kh);
        acc0 = __builtin_amdgcn_wmma_f32_16x16x4_f32(false, a, false, b0, (short)0, acc0, false, false);
        acc1 = __builtin_amdgcn_wmma_f32_16x16x4_f32(false, a, false, b1, (short)0, acc1, false, false);
        acc2 = __builtin_amdgcn_wmma_f32_16x16x4_f32(false, a, false, b2, (short)0, acc2, false, false);
        acc3 = __builtin_amdgcn_wmma_f32_16x16x4_f32(false, a, false, b3, (short)0, acc3, false, false);
    }

    // D layout: element v -> M = m0 + v + 8*(lane>=16), N = n0 + nt*16 + (lane&15)
    const int hi = lane >> 4;
    #pragma unroll
    for (int v = 0; v < 8; ++v) {
        const int   mrel = jrow + v + 8 * hi;
        const float bb   = bvec[mrel];
        float* o = xz + (size_t)(m0 + v + 8 * hi) * TSTEPS + n0 + ml;
        o[ 0] = acc0[v] + bb;
        o[16] = acc1[v] + bb;
        o[32] = acc2[v] + bb;
        o[48] = acc3[v] + bb;
    }
}

// -----------------------------------------------------------------------------
// Kernel B: one LSTM step. Wave w owns output index j: computes the 4 gate-row
// dot products Wh[g*2048+j][:] . h_{t-1} (h staged in LDS per block), adds the
// precomputed x-contribution, applies gates, updates c in place and writes h_t.
// Bandwidth-bound on the 64 MB/step of Wh traffic, which lives in L2 (192 MB).
// -----------------------------------------------------------------------------
__global__ __launch_bounds__(256)
void lstm_step(const float* __restrict__ Wi, const float* __restrict__ Wf,
               const float* __restrict__ Wo, const float* __restrict__ Wg,
               const float* __restrict__ xz,      // [8192, 1024]
               const float* __restrict__ h_prev,  // [2048]
               float* __restrict__ c,             // [2048], updated in place
               float* __restrict__ h_out,         // [2048]
               int t)
{
    __shared__ float hs[HIDD];
    for (int i = threadIdx.x; i < HIDD; i += 256) hs[i] = h_prev[i];
    __syncthreads();

    const int lane = threadIdx.x & 31;
    const int j    = blockIdx.x * 8 + (threadIdx.x >> 5); // 0..2047

    const float* rw0 = Wi + (size_t)j * KAUG + INPD;  // h-part = cols [1024,3072)
    const float* rw1 = Wf + (size_t)j * KAUG + INPD;
    const float* rw2 = Wo + (size_t)j * KAUG + INPD;
    const float* rw3 = Wg + (size_t)j * KAUG + INPD;

    float dot[4];
    const float* rows[4] = { rw0, rw1, rw2, rw3 };
    #pragma unroll
    for (int g = 0; g < 4; ++g) {
        const float4* wr = (const float4*)rows[g];
        const float4* hr = (const float4*)hs;
        float s = 0.f;
        #pragma unroll 4
        for (int k4 = lane; k4 < HIDD / 4; k4 += 32) {   // b128 loads
            float4 w = wr[k4];
            float4 h = hr[k4];
            s += w.x * h.x + w.y * h.y + w.z * h.z + w.w * h.w;
        }
        #pragma unroll
        for (int off = 16; off > 0; off >>= 1) s += __shfl_xor(s, off, 32);
        dot[g] = s;
    }

    if (lane == 0) {
        const size_t jt = (size_t)j * TSTEPS + t;
        const float zi = dot[0] + xz[jt];
        const float zf = dot[1] + xz[(size_t)1 * HIDD * TSTEPS + jt];
        const float zo = dot[2] + xz[(size_t)2 * HIDD * TSTEPS + jt];
        const float zg = dot[3] + xz[(size_t)3 * HIDD * TSTEPS + jt];
        const float ii = 1.f / (1.f + expf(-zi));
        const float ff = 1.f / (1.f + expf(-zf));
        const float oo = 1.f / (1.f + expf(-zo));
        const float gg = tanhf(zg);
        const float cn = ff * c[j] + ii * gg;
        c[j]     = cn;
        h_out[j] = oo * tanhf(cn);
    }
}

__global__ void lstm_init_c(const float* __restrict__ c0, float* __restrict__ c) {
    int i = blockIdx.x * blockDim.x + threadIdx.x;
    if (i < HIDD) c[i] = c0[i];
}

// Append h_T (copy of last outs row) and c_T to d_out tail.
__global__ void lstm_finish(float* __restrict__ out, const float* __restrict__ c) {
    int i = blockIdx.x * blockDim.x + threadIdx.x;
    if (i < HIDD) {
        out[(size_t)TSTEPS * HIDD + i]        = out[(size_t)(TSTEPS - 1) * HIDD + i];
        out[(size_t)TSTEPS * HIDD + HIDD + i] = c[i];
    }
}

extern "C" void kernel_launch(void* const* d_in, const int* in_sizes, int n_in,
                              void* d_out, int out_size, void* d_ws, size_t ws_size,
                              hipStream_t stream) {
    (void)in_sizes; (void)n_in; (void)out_size; (void)ws_size;
    const float* x  = (const float*)d_in[0];
    const float* h0 = (const float*)d_in[1];
    const float* c0 = (const float*)d_in[2];
    const float* Wi = (const float*)d_in[3]; const float* bi = (const float*)d_in[4];
    const float* Wf = (const float*)d_in[5]; const float* bf = (const float*)d_in[6];
    const float* Wo = (const float*)d_in[7]; const float* bo = (const float*)d_in[8];
    const float* Wg = (const float*)d_in[9]; const float* bg = (const float*)d_in[10];

    float* out  = (float*)d_out;
    float* xz   = (float*)d_ws;                          // 8192*1024 f32 = 32 MB
    float* cbuf = xz + (size_t)ROWS4H * TSTEPS;          // 2048 f32

    // Precompute x-contribution for all timesteps (WMMA fp32 GEMM), incl. bias.
    lstm_xgemm_wmma<<<1024, 256, 0, stream>>>(x, Wi, Wf, Wo, Wg, bi, bf, bo, bg, xz);
    lstm_init_c<<<8, 256, 0, stream>>>(c0, cbuf);

    // Sequential recurrence: one fused GEMV+gates kernel per step; h history
    // lives in d_out, so step t reads the row step t-1 wrote.
    for (int t = 0; t < TSTEPS; ++t) {
        const float* hp = (t == 0) ? h0 : out + (size_t)(t - 1) * HIDD;
        lstm_step<<<256, 256, 0, stream>>>(Wi, Wf, Wo, Wg, xz, hp, cbuf,
                                           out + (size_t)t * HIDD, t);
    }
    lstm_finish<<<8, 256, 0, stream>>>(out, cbuf);
}